// RelGraphConv_8263517078052
// MI455X (gfx1250) — compile-verified
//
#include <hip/hip_runtime.h>
#include <hip/hip_bf16.h>

// R-GCN on MI455X (gfx1250, wave32, WMMA).
//
// Strategy (see analysis): counting-sort edges by relation into 128-aligned
// buckets, then edge-centric f32 WMMA gather-GEMM-scatter that keeps all hot
// data (feat 25.6MB, out 25.6MB, W 256KB) in the 192MB L2 instead of
// materializing the reference's 410MB xproj tensor in HBM.

typedef float v2f __attribute__((ext_vector_type(2)));
typedef float v8f __attribute__((ext_vector_type(8)));

#define IN_F 64
#define OUT_F 64
#define LDS_STRIDE 68   // 64 + 4 pad: conflict-free ds_load_b64 B-fragments

static __device__ __forceinline__ v8f wmma_f32_k4(v2f a, v2f b, v8f c) {
  // D = A(16x4,f32) * B(4x16,f32) + C(16x16,f32)
  return __builtin_amdgcn_wmma_f32_16x16x4_f32(
      /*neg_a=*/false, a, /*neg_b=*/false, b,
      /*c_mod=*/(short)0, c, /*reuse_a=*/false, /*reuse_b=*/false);
}

static __device__ __forceinline__ void atomic_add_f32(float* p, float v) {
  __hip_atomic_fetch_add(p, v, __ATOMIC_RELAXED, __HIP_MEMORY_SCOPE_AGENT);
}

// ---------------- counting sort by etype ----------------

__global__ void k_zero(unsigned* cnt) {
  if (threadIdx.x < 16) cnt[threadIdx.x] = 0u;
}

__global__ void k_count(const int* __restrict__ etype, unsigned* cnt, int E) {
  int e = blockIdx.x * 256 + threadIdx.x;
  if (e < E) atomicAdd(&cnt[etype[e]], 1u);
}

__global__ void k_scan(const unsigned* __restrict__ cnt, unsigned* offs,
                       unsigned* cur, int R) {
  if (threadIdx.x == 0) {
    unsigned o = 0;
    for (int r = 0; r < R; ++r) {
      offs[r] = o;
      cur[r]  = o;
      o += (cnt[r] + 127u) & ~127u;  // pad each bucket to 128 edges
    }
    offs[R] = o;
  }
}

__global__ void k_fill(int* sorted, int cap) {
  int i = blockIdx.x * 256 + threadIdx.x;
  if (i < cap) sorted[i] = -1;  // -1 = padding slot
}

__global__ void k_scatter(const int* __restrict__ etype, unsigned* cur,
                          int* sorted, int E) {
  int e = blockIdx.x * 256 + threadIdx.x;
  if (e < E) {
    unsigned p = atomicAdd(&cur[etype[e]], 1u);
    sorted[p] = e;
  }
}

// ---------------- self-loop: out = feat @ loop_weight + bias ----------------

__global__ __launch_bounds__(256) void k_selfloop(
    const float* __restrict__ feat, const float* __restrict__ lw,
    const float* __restrict__ bias, float* __restrict__ out, int N) {
  __shared__ float wlds[64 * LDS_STRIDE];
  const int tid = threadIdx.x;
  // store W^T into LDS: wlds[col*STRIDE + row] = W[row][col]
  for (int i = tid; i < IN_F * OUT_F; i += 256) {
    int row = i >> 6, col = i & 63;
    wlds[col * LDS_STRIDE + row] = lw[i];
  }
  __syncthreads();

  const int lane  = tid & 31, w = tid >> 5;
  const int khalf = lane >> 4, m = lane & 15;
  const int rowBase = blockIdx.x * 128 + w * 16;

  int ar = rowBase + m;
  if (ar >= N) ar = N - 1;  // clamp; OOB rows discarded at store
  const float* arow = feat + (size_t)ar * IN_F + 2 * khalf;

  v8f acc0 = {}, acc1 = {}, acc2 = {}, acc3 = {};
#pragma unroll
  for (int k = 0; k < 16; ++k) {
    v2f a = *(const v2f*)(arow + 4 * k);
    int brow = 4 * k + 2 * khalf;
    v2f b0 = *(const v2f*)&wlds[(m + 0)  * LDS_STRIDE + brow];
    v2f b1 = *(const v2f*)&wlds[(m + 16) * LDS_STRIDE + brow];
    v2f b2 = *(const v2f*)&wlds[(m + 32) * LDS_STRIDE + brow];
    v2f b3 = *(const v2f*)&wlds[(m + 48) * LDS_STRIDE + brow];
    acc0 = wmma_f32_k4(a, b0, acc0);
    acc1 = wmma_f32_k4(a, b1, acc1);
    acc2 = wmma_f32_k4(a, b2, acc2);
    acc3 = wmma_f32_k4(a, b3, acc3);
  }

  float bb0 = bias[m +  0], bb1 = bias[m + 16];
  float bb2 = bias[m + 32], bb3 = bias[m + 48];
#pragma unroll
  for (int j = 0; j < 8; ++j) {
    int grow = rowBase + j + 8 * khalf;  // C/D layout: vgpr j, half -> row
    if (grow < N) {
      float* o = out + (size_t)grow * OUT_F + m;
      o[0]  = acc0[j] + bb0;
      o[16] = acc1[j] + bb1;
      o[32] = acc2[j] + bb2;
      o[48] = acc3[j] + bb3;
    }
  }
}

// -------- edge messages: out[dst] += feat[src] @ W[etype] (bucketed) --------

__global__ __launch_bounds__(256) void k_msgs(
    const float* __restrict__ feat, const float* __restrict__ weight,
    const int* __restrict__ src, const int* __restrict__ dst,
    const unsigned* __restrict__ offs, const int* __restrict__ sorted,
    float* __restrict__ out, int R) {
  __shared__ float wlds[64 * LDS_STRIDE];
  const int tid = threadIdx.x;
  const unsigned blockBase = blockIdx.x * 128u;
  const unsigned total = offs[R];
  if (blockBase >= total) return;  // uniform exit

  // which relation owns this 128-edge (single-relation, padded) block
  int r = 0;
  for (int i = 1; i < R; ++i) r += (blockBase >= offs[i]) ? 1 : 0;

  const float* W = weight + (size_t)r * (IN_F * OUT_F);
  for (int i = tid; i < IN_F * OUT_F; i += 256) {
    int row = i >> 6, col = i & 63;
    wlds[col * LDS_STRIDE + row] = W[i];
  }
  __syncthreads();

  const int lane  = tid & 31, w = tid >> 5;
  const int khalf = lane >> 4, m = lane & 15;
  const int tileBase = (int)blockBase + w * 16;

  // A-row gather: lane's edge m; padding rows read row 0 (result discarded)
  int e_m  = sorted[tileBase + m];
  int srow = (e_m >= 0) ? src[e_m] : 0;
  const float* arow = feat + (size_t)srow * IN_F + 2 * khalf;

  v8f acc0 = {}, acc1 = {}, acc2 = {}, acc3 = {};
#pragma unroll
  for (int k = 0; k < 16; ++k) {
    v2f a = *(const v2f*)(arow + 4 * k);
    int brow = 4 * k + 2 * khalf;
    v2f b0 = *(const v2f*)&wlds[(m + 0)  * LDS_STRIDE + brow];
    v2f b1 = *(const v2f*)&wlds[(m + 16) * LDS_STRIDE + brow];
    v2f b2 = *(const v2f*)&wlds[(m + 32) * LDS_STRIDE + brow];
    v2f b3 = *(const v2f*)&wlds[(m + 48) * LDS_STRIDE + brow];
    acc0 = wmma_f32_k4(a, b0, acc0);
    acc1 = wmma_f32_k4(a, b1, acc1);
    acc2 = wmma_f32_k4(a, b2, acc2);
    acc3 = wmma_f32_k4(a, b3, acc3);
  }

  // scatter-add: row j+8*khalf of D belongs to edge sorted[tileBase+row]
#pragma unroll
  for (int j = 0; j < 8; ++j) {
    int rowm = j + 8 * khalf;
    int e = sorted[tileBase + rowm];
    if (e >= 0) {
      int d = dst[e];
      float* o = out + (size_t)d * OUT_F + m;
      atomic_add_f32(o + 0,  acc0[j]);
      atomic_add_f32(o + 16, acc1[j]);
      atomic_add_f32(o + 32, acc2[j]);
      atomic_add_f32(o + 48, acc3[j]);
    }
  }
}

// ---------------- launch ----------------

extern "C" void kernel_launch(void* const* d_in, const int* in_sizes, int n_in,
                              void* d_out, int out_size, void* d_ws, size_t ws_size,
                              hipStream_t stream) {
  const float* feat   = (const float*)d_in[0];
  const float* weight = (const float*)d_in[1];
  const float* h_bias = (const float*)d_in[2];
  const float* loop_w = (const float*)d_in[3];
  const int*   src    = (const int*)d_in[4];
  const int*   dst    = (const int*)d_in[5];
  const int*   etype  = (const int*)d_in[6];
  float* out = (float*)d_out;

  const int N = in_sizes[0] / IN_F;
  const int E = in_sizes[4];
  const int R = in_sizes[1] / (IN_F * OUT_F);

  unsigned char* ws = (unsigned char*)d_ws;
  unsigned* cnt  = (unsigned*)(ws);         // [16]
  unsigned* offs = (unsigned*)(ws + 64);    // [R+1]
  unsigned* cur  = (unsigned*)(ws + 192);   // [16]
  int* sorted    = (int*)(ws + 256);        // [E + R*128] padded sort buffer
  const int cap  = E + R * 128;

  k_zero<<<1, 32, 0, stream>>>(cnt);
  k_count<<<(E + 255) / 256, 256, 0, stream>>>(etype, cnt, E);
  k_scan<<<1, 32, 0, stream>>>(cnt, offs, cur, R);
  k_fill<<<(cap + 255) / 256, 256, 0, stream>>>(sorted, cap);
  k_scatter<<<(E + 255) / 256, 256, 0, stream>>>(etype, cur, sorted, E);

  // init out with self-loop + bias, then atomically accumulate messages
  k_selfloop<<<(N + 127) / 128, 256, 0, stream>>>(feat, loop_w, h_bias, out, N);
  k_msgs<<<(E + 127) / 128 + R, 256, 0, stream>>>(feat, weight, src, dst,
                                                  offs, sorted, out, R);
}